// LGConv_4492535791995
// MI455X (gfx1250) — compile-verified
//
#include <hip/hip_runtime.h>

#define F 128

typedef float v2f __attribute__((ext_vector_type(2)));
typedef float v8f __attribute__((ext_vector_type(8)));

// ---------------- helper kernels ----------------

__global__ void zero_f32(float* p, int n) {
    int i = blockIdx.x * blockDim.x + threadIdx.x;
    if (i < n) p[i] = 0.0f;
}

__global__ void degree_kernel(const int* __restrict__ dst, float* __restrict__ deg, int E) {
    int e = blockIdx.x * blockDim.x + threadIdx.x;
    if (e < E) unsafeAtomicAdd(&deg[dst[e]], 1.0f);
}

__global__ void norm_kernel(float* __restrict__ norm, int n) {
    int i = blockIdx.x * blockDim.x + threadIdx.x;
    if (i < n) {
        float d = norm[i];
        d = d < 1.0f ? 1.0f : d;
        norm[i] = rsqrtf(d);
    }
}

// C = alpha0 * feat ; B0 = feat * norm ; B1 = 0
__global__ void init_kernel(const float* __restrict__ feat, const float* __restrict__ norm,
                            const float* __restrict__ alpha,
                            float* __restrict__ C, float* __restrict__ B0,
                            float* __restrict__ B1, int total) {
    int i = blockIdx.x * blockDim.x + threadIdx.x;
    if (i >= total) return;
    float f = feat[i];
    int row = i >> 7;
    C[i]  = alpha[0] * f;
    B0[i] = f * norm[row];
    B1[i] = 0.0f;
}

// one wave (32 lanes) per edge; each lane moves a float4 (128 floats per edge)
__global__ void spmm_kernel(const int* __restrict__ src, const int* __restrict__ dst,
                            const float* __restrict__ B0, float* __restrict__ B1, int E) {
    int gid  = blockIdx.x * blockDim.x + threadIdx.x;
    int e    = gid >> 5;
    int lane = gid & 31;
    if (e >= E) return;
    int s = src[e];
    int d = dst[e];
    const float4* s4 = (const float4*)(B0 + (size_t)s * F);
    float4 v = s4[lane];
    float* out = B1 + (size_t)d * F + lane * 4;
    unsafeAtomicAdd(out + 0, v.x);
    unsafeAtomicAdd(out + 1, v.y);
    unsafeAtomicAdd(out + 2, v.z);
    unsafeAtomicAdd(out + 3, v.w);
}

// h = B1*norm ; C += alpha[ai]*h ; if NEXT: B0 = h*norm, B1 = 0 (for next round / next replay)
template <bool NEXT>
__global__ void post_kernel(float* __restrict__ B1, const float* __restrict__ norm,
                            const float* __restrict__ alpha, int ai,
                            float* __restrict__ C, float* __restrict__ B0, int total) {
    int i = blockIdx.x * blockDim.x + threadIdx.x;
    if (i >= total) return;
    int row = i >> 7;
    float nv = norm[row];
    float h = B1[i] * nv;
    C[i] += alpha[ai] * h;
    if (NEXT) {
        B0[i] = h * nv;
        B1[i] = 0.0f;
    }
}

// ---------------- WMMA GEMM: out = C @ W^T + 3*b, in place on C (= d_out) ----------------
// Block: 256 threads = 8 waves. Each wave: 16 rows x 128 cols, K=128 in steps of 4.
// W staged in LDS paired-transposed:
//   ldsP[(k>>1)*256 + col*2 + (k&1)] = W[col][k]
// so each B fragment (W^T[kk][col], W^T[kk+1][col]) is one contiguous, 8-byte-aligned
// ds_load_b64. Consecutive 8B global chunks of W map to contiguous 8B LDS chunks,
// enabling GLOBAL_LOAD_ASYNC_TO_LDS_B64 staging where the toolchain supports it.

#if defined(__has_builtin)
#if __has_builtin(__builtin_amdgcn_global_load_async_to_lds_b64) && \
    __has_builtin(__builtin_amdgcn_s_wait_asynccnt)
#define LGC_ASYNC_STAGE 1
#endif
#endif

#ifdef LGC_ASYNC_STAGE
typedef __attribute__((__vector_size__(8))) int lgc_v2i;
typedef __attribute__((address_space(1))) lgc_v2i* lgc_gptr;
typedef __attribute__((address_space(3))) lgc_v2i* lgc_lptr;
#endif

__global__ __launch_bounds__(256) void gemm_kernel(float* __restrict__ C,
                                                   const float* __restrict__ W,
                                                   const float* __restrict__ bias, int n) {
    __shared__ float ldsP[F * F]; // 64 KB of the 320 KB WGP LDS

    int tid = threadIdx.x;
    // pair p = o*64 + j  maps  W[o][2j..2j+1] (global) -> ldsP[j*256 + 2o .. +1]
#ifdef LGC_ASYNC_STAGE
    for (int p = tid; p < (F * F) / 2; p += 256) {
        int o = p >> 6;
        int j = p & 63;
        __builtin_amdgcn_global_load_async_to_lds_b64(
            (lgc_gptr)(W + o * F + 2 * j),
            (lgc_lptr)(ldsP + j * (2 * F) + 2 * o), 0, 0);
    }
    __builtin_amdgcn_s_wait_asynccnt(0);
#else
    for (int p = tid; p < (F * F) / 2; p += 256) {
        int o = p >> 6;
        int j = p & 63;
        float2 w2 = *(const float2*)(W + o * F + 2 * j);
        *(float2*)(ldsP + j * (2 * F) + 2 * o) = w2;
    }
#endif
    __syncthreads();

    int wave = tid >> 5;
    int lane = tid & 31;
    int half = lane >> 4;   // selects K pair within fragment
    int ln   = lane & 15;
    int rowBase = blockIdx.x * 128 + wave * 16;

    // accumulators: 8 col-tiles of 16; init with 3*b (bias added K+1 = 3 times)
    v8f acc[8];
#pragma unroll
    for (int c = 0; c < 8; ++c) {
        float bv = 3.0f * bias[c * 16 + ln];
#pragma unroll
        for (int r = 0; r < 8; ++r) acc[c][r] = bv;
    }

    // A: lane L supplies row M = L&15; an invalid row only pollutes discarded D rows,
    // so clamp instead of branching -> unconditional global_load_b64 in the hot loop.
    int arow = rowBase + ln;
    if (arow > n - 1) arow = n - 1;
    const float2* Arow = (const float2*)(C + (size_t)arow * F);

    for (int k2 = 0; k2 < F / 2; k2 += 2) { // k2 = k/2, K-step of 4
        float2 av = Arow[k2 + half];        // A[row][k+2h .. k+2h+1]
        v2f a;
        a.x = av.x;
        a.y = av.y;
        const float* brow = ldsP + (size_t)(k2 + half) * (2 * F);
#pragma unroll
        for (int c = 0; c < 8; ++c) {
            int col = c * 16 + ln;
            float2 bv2 = *(const float2*)(brow + 2 * col); // W^T[kk][col], W^T[kk+1][col]
            v2f bf;
            bf.x = bv2.x;
            bf.y = bv2.y;
            acc[c] = __builtin_amdgcn_wmma_f32_16x16x4_f32(
                false, a, false, bf, (short)0, acc[c], false, false);
        }
    }

    // store D: VGPR r -> row rowBase + r + 8*half, col = c*16 + ln
#pragma unroll
    for (int c = 0; c < 8; ++c) {
        int col = c * 16 + ln;
#pragma unroll
        for (int r = 0; r < 8; ++r) {
            int row = rowBase + r + 8 * half;
            if (row < n) C[(size_t)row * F + col] = acc[c][r];
        }
    }
}

// ---------------- launcher ----------------

extern "C" void kernel_launch(void* const* d_in, const int* in_sizes, int n_in,
                              void* d_out, int out_size, void* d_ws, size_t ws_size,
                              hipStream_t stream) {
    const float* feat  = (const float*)d_in[0];
    const int*   src   = (const int*)d_in[1];
    const int*   dst   = (const int*)d_in[2];
    const float* W     = (const float*)d_in[3];
    const float* bias  = (const float*)d_in[4];
    const float* alpha = (const float*)d_in[5];

    const int N = in_sizes[0] / F;
    const int E = in_sizes[1];
    const int total = N * F;

    float* C    = (float*)d_out;               // combined features, then final result (in place)
    float* norm = (float*)d_ws;                // N floats (degree, then norm)
    float* B0   = norm + (((size_t)N + 255) & ~(size_t)255);
    float* B1   = B0 + (size_t)total;

    const int t = 256;
    zero_f32<<<(N + t - 1) / t, t, 0, stream>>>(norm, N);
    degree_kernel<<<(E + t - 1) / t, t, 0, stream>>>(dst, norm, E);
    norm_kernel<<<(N + t - 1) / t, t, 0, stream>>>(norm, N);

    init_kernel<<<(total + t - 1) / t, t, 0, stream>>>(feat, norm, alpha, C, B0, B1, total);

    long sThreads = (long)E * 32;
    // round 1
    spmm_kernel<<<(int)((sThreads + t - 1) / t), t, 0, stream>>>(src, dst, B0, B1, E);
    post_kernel<true><<<(total + t - 1) / t, t, 0, stream>>>(B1, norm, alpha, 1, C, B0, total);
    // round 2
    spmm_kernel<<<(int)((sThreads + t - 1) / t), t, 0, stream>>>(src, dst, B0, B1, E);
    post_kernel<false><<<(total + t - 1) / t, t, 0, stream>>>(B1, norm, alpha, 2, C, nullptr, total);

    // final GEMM with fp32 WMMA, in place on d_out
    gemm_kernel<<<(N + 127) / 128, 256, 0, stream>>>(C, W, bias, N);
}